// CLRNetAssign_8074538517113
// MI455X (gfx1250) — compile-verified
//
#include <hip/hip_runtime.h>
#include <hip/hip_bf16.h>

#define BB 8
#define NN 4096
#define MM 32
#define KK 72
#define DD 78
#define W_REG 3.0f
#define W_CLS 1.0f
#define IOU_L 15.0f
#define EPSF 1e-12f
#define BIGF 100000.0f

typedef __attribute__((ext_vector_type(2))) float v2f;
typedef __attribute__((ext_vector_type(8))) float v8f;
typedef __attribute__((ext_vector_type(4))) unsigned int u32x4;
typedef __attribute__((ext_vector_type(4))) int i32x4;
typedef __attribute__((ext_vector_type(8))) int i32x8;

__device__ __forceinline__ float focal_diff(float x) {
    float p = 1.0f / (1.0f + expf(-x));
    float pos = -logf(p + EPSF) * 0.25f * (1.0f - p) * (1.0f - p);
    float neg = -logf(1.0f - p + EPSF) * 0.75f * p * p;
    return pos - neg;
}

// ---------------- init: zero match-mask & maxima, compute per-target valid counts
__global__ void init_kernel(const float* __restrict__ tgt, const int* __restrict__ imgw,
                            int* __restrict__ LEN, int* __restrict__ MAXD,
                            int* __restrict__ MAXXY, int* __restrict__ MAXTH,
                            int* __restrict__ MATCH) {
    int idx = blockIdx.x * 256 + threadIdx.x;
    if (idx < BB * NN) MATCH[idx] = 0;
    if (idx < BB) { MAXD[idx] = 0; MAXXY[idx] = 0; MAXTH[idx] = 0; }
    if (idx < BB * MM) {
        float fw = (float)imgw[0];
        const float* tb = tgt + (size_t)idx * DD;
        int c = 0;
        for (int k = 0; k < KK; k++) {
            float t = tb[6 + k];
            c += (t >= 0.0f && t < fw) ? 1 : 0;
        }
        LEN[idx] = c;
    }
}

// ---------------- pass A: S = masked L1 over K (targets staged by TDM),
//                  xysq via WMMA gram trick, batch maxima
__global__ void __launch_bounds__(64) passA_kernel(
    const float* __restrict__ preds, const float* __restrict__ tgt,
    const int* __restrict__ LEN, const int* __restrict__ imgw,
    float* __restrict__ S, float* __restrict__ XYSQ,
    int* __restrict__ MAXD, int* __restrict__ MAXXY, int* __restrict__ MAXTH) {
    __shared__ float tls[MM][KK];
    __shared__ float vms[MM][KK];
    __shared__ float bxy[MM][4];
    __shared__ float tths[MM];
    __shared__ float pxs[64], pys[64];
    __shared__ float red[3][64];

    int b = blockIdx.y;
    int n0 = blockIdx.x * 64;
    int tid = threadIdx.x;
    int lane = tid & 31;
    int wid  = tid >> 5;
    float fw = (float)imgw[0];

    // --- TDM: DMA the 32x72 (row-stride 78) target-x tile into packed LDS tls[][]
    if (wid == 0) {
        unsigned long long gaddr =
            (unsigned long long)(const void*)(tgt + (size_t)b * MM * DD + 6);
        unsigned int ldsoff =
            (unsigned int)(unsigned long long)(const void*)&tls[0][0];
        u32x4 g0;
        g0[0] = 1u;                                   // count=1, user descriptor
        g0[1] = ldsoff;                               // lds_addr (bytes)
        g0[2] = (unsigned int)(gaddr & 0xffffffffull);        // global_addr[31:0]
        g0[3] = (unsigned int)((gaddr >> 32) & 0x01ffffffull) // global_addr[56:32]
                | 0x80000000u;                        // type=2 ("image")
        i32x8 g1;
        g1[0] = 0x00020000;          // wg_mask=0, data_size=2 (4B), no pad/iter
        g1[1] = (KK & 0xffff) << 16; // tensor_dim0[15:0]  (=72)
        g1[2] = (KK >> 16) | ((MM & 0xffff) << 16); // dim0 hi | tensor_dim1 lo (=32)
        g1[3] = (MM >> 16) | ((KK & 0xffff) << 16); // dim1 hi | tile_dim0 (=72)
        g1[4] = MM;                  // tile_dim1 (=32), tile_dim2=0 -> 2D
        g1[5] = DD;                  // tensor_dim0_stride[31:0] (=78 elements)
        g1[6] = 0;                   // stride0 hi | tensor_dim1_stride lo
        g1[7] = 0;                   // tensor_dim1_stride hi
        i32x4 g2 = {0, 0, 0, 0};     // tensor_dim2=0, tile_dim3=0 -> unused
        i32x4 g3 = {0, 0, 0, 0};
        i32x8 g4 = {0, 0, 0, 0, 0, 0, 0, 0};  // extra descriptor words (unused)
        __builtin_amdgcn_tensor_load_to_lds(g0, g1, g2, g3, g4, 0);
    }
    if (tid < MM) {
        const float* tb = tgt + (size_t)(b * MM + tid) * DD;
        float tx = tb[2], ty = tb[3];
        bxy[tid][0] = -2.0f * tx;
        bxy[tid][1] = -2.0f * ty;
        bxy[tid][2] = 1.0f;
        bxy[tid][3] = tx * tx + ty * ty;
        tths[tid] = tb[4];
    }
    if (wid == 0) __builtin_amdgcn_s_wait_tensorcnt((short)0);
    __syncthreads();
    // derive valid mask from landed tile
    for (int i = tid; i < MM * KK; i += 64) {
        float t = (&tls[0][0])[i];
        (&vms[0][0])[i] = (t >= 0.0f && t < fw) ? 1.0f : 0.0f;
    }
    __syncthreads();

    int n = n0 + tid;
    const float* prow = preds + (size_t)(b * NN + n) * DD;
    __builtin_prefetch(prow, 0, 1);                 // global_prefetch_b8
    float pk[KK];
    for (int k = 0; k < KK; k++) pk[k] = prow[6 + k];
    float px = prow[2], py = prow[3], pth = prow[4];
    pxs[tid] = px; pys[tid] = py;

    float dmax = 0.0f, thmax = 0.0f;
    for (int m = 0; m < MM; m++) {
        float s = 0.0f;
        for (int k = 0; k < KK; k++) s += vms[m][k] * fabsf(pk[k] - tls[m][k]);
        S[(size_t)(b * MM + m) * NN + n] = s;
        int len = LEN[b * MM + m];
        float lf = (float)(len > 1 ? len : 1);
        dmax = fmaxf(dmax, s / lf);
        thmax = fmaxf(thmax, fabsf(pth - tths[m]));
    }
    __syncthreads();

    // WMMA: pairwise squared xy distance, 16x16 tiles, K=4 gram trick.
    // A row n: (px, py, px^2+py^2, 1) ; B col m: (-2tx, -2ty, 1, tx^2+ty^2)
    int r  = lane & 15;
    int hi = lane >> 4;   // 0: K=0,1  1: K=2,3 (32-bit 16x4 operand layout)
    float xymax = 0.0f;
    for (int s2 = 0; s2 < 2; s2++) {
        int rowbase = wid * 32 + s2 * 16;
        float apx = pxs[rowbase + r], apy = pys[rowbase + r];
        v2f a;
        a.x = hi ? (apx * apx + apy * apy) : apx;
        a.y = hi ? 1.0f : apy;
        for (int mt = 0; mt < 2; mt++) {
            int mc = mt * 16 + r;
            v2f bv;
            bv.x = hi ? bxy[mc][2] : bxy[mc][0];
            bv.y = hi ? bxy[mc][3] : bxy[mc][1];
            v8f c = {};
            c = __builtin_amdgcn_wmma_f32_16x16x4_f32(
                    false, a, false, bv, (short)0, c, false, false);
#pragma unroll
            for (int i = 0; i < 8; i++) {
                int row = i + hi * 8;
                int n2 = n0 + rowbase + row;
                float val = c[i];
                XYSQ[(size_t)(b * MM + mc) * NN + n2] = val;
                xymax = fmaxf(xymax, sqrtf(fmaxf(val, 0.0f)));
            }
        }
    }

    red[0][tid] = dmax; red[1][tid] = xymax; red[2][tid] = thmax;
    __syncthreads();
    for (int h = 32; h > 0; h >>= 1) {
        if (tid < h)
            for (int q = 0; q < 3; q++)
                red[q][tid] = fmaxf(red[q][tid], red[q][tid + h]);
        __syncthreads();
    }
    if (tid == 0) {   // all values >= 0 -> int compare == float compare
        atomicMax(&MAXD[b],  __float_as_int(red[0][0]));
        atomicMax(&MAXXY[b], __float_as_int(red[1][0]));
        atomicMax(&MAXTH[b], __float_as_int(red[2][0]));
    }
}

// ---------------- pass B: cost + iou from S/xysq and batch maxima
__global__ void __launch_bounds__(256) passB_kernel(
    const float* __restrict__ preds, const float* __restrict__ tgt,
    const int* __restrict__ masks, const int* __restrict__ LEN,
    const int* __restrict__ MAXD, const int* __restrict__ MAXXY, const int* __restrict__ MAXTH,
    const float* __restrict__ S, const float* __restrict__ XYSQ,
    float* __restrict__ COST, float* __restrict__ IOUW) {
    int b = blockIdx.y;
    int n = blockIdx.x * 256 + threadIdx.x;
    const float* prow = preds + (size_t)(b * NN + n) * DD;
    float d0 = focal_diff(prow[0]);
    float d1 = focal_diff(prow[1]);
    float pth = prow[4];
    float maxd  = fmaxf(__int_as_float(MAXD[b]),  1e-6f);
    float maxxy = fmaxf(__int_as_float(MAXXY[b]), 1e-6f);
    float maxth = fmaxf(__int_as_float(MAXTH[b]), 1e-6f);
    for (int m = 0; m < MM; m++) {
        const float* tb = tgt + (size_t)(b * MM + m) * DD;
        size_t off = (size_t)(b * MM + m) * NN + n;
        float s = S[off];
        float xysq = XYSQ[off];
        int len = LEN[b * MM + m];
        float dist = s / (float)(len > 1 ? len : 1);
        float dist_s = 1.0f - dist / maxd + 0.01f;
        float xy = sqrtf(fmaxf(xysq, 0.0f));
        float xy_s = 1.0f - xy / maxxy + 0.01f;
        float th = fabsf(pth - tb[4]);
        float th_s = 1.0f - th / maxth + 0.01f;
        int label = (int)tb[1];
        float cls = label ? d1 : d0;
        float prod = dist_s * xy_s * th_s;
        float cost = -prod * prod * W_REG + cls * W_CLS;
        if (masks[b * MM + m] == 0) cost = BIGF;
        COST[off] = cost;
        float fl = (float)len;
        IOUW[off] = (2.0f * IOU_L * fl - s) / (2.0f * IOU_L * fl + s + 1e-9f);
    }
}

// ---------------- pass C: per-(b,m) top4 iou (dynamic_k) + top4 lowest cost -> bitmask scatter
__global__ void __launch_bounds__(128) topk_kernel(
    const int* __restrict__ masks, const float* __restrict__ COST,
    const float* __restrict__ IOUW, int* __restrict__ MATCH) {
    int bm = blockIdx.x;
    int b = bm / MM, m = bm % MM;
    if (masks[bm] == 0) return;   // masked targets: matching zeroed anyway
    int tid = threadIdx.x;

    __shared__ float sli[128][4];
    __shared__ float slc[128][4];
    __shared__ int   sci[128][4];

    float iv[4], cv[4]; int ci[4];
    for (int j = 0; j < 4; j++) { iv[j] = -1.0f; cv[j] = 3.0e38f; ci[j] = 0x7fffffff; }

    const float* crow = COST + (size_t)bm * NN;
    const float* irow = IOUW + (size_t)bm * NN;
    for (int n = tid; n < NN; n += 128) {
        float io = fmaxf(irow[n], 0.0f);
        if (io > iv[3]) {
            int j = 3;
            while (j > 0 && io > iv[j - 1]) { iv[j] = iv[j - 1]; j--; }
            iv[j] = io;
        }
        float c = crow[n];
        if (c < cv[3] || (c == cv[3] && n < ci[3])) {
            int j = 3;
            while (j > 0 && (c < cv[j - 1] || (c == cv[j - 1] && n < ci[j - 1]))) {
                cv[j] = cv[j - 1]; ci[j] = ci[j - 1]; j--;
            }
            cv[j] = c; ci[j] = n;
        }
    }
    for (int j = 0; j < 4; j++) { sli[tid][j] = iv[j]; slc[tid][j] = cv[j]; sci[tid][j] = ci[j]; }
    __syncthreads();
    for (int h = 64; h > 0; h >>= 1) {
        if (tid < h) {
            float a0[4], b0[4], a1[4], b1[4]; int ai[4], bi[4];
            for (int j = 0; j < 4; j++) {
                a0[j] = sli[tid][j];     b0[j] = sli[tid + h][j];
                a1[j] = slc[tid][j];     b1[j] = slc[tid + h][j];
                ai[j] = sci[tid][j];     bi[j] = sci[tid + h][j];
            }
            float r0[4], r1[4]; int ri[4];
            int pa = 0, pb = 0;
            for (int o = 0; o < 4; o++) {            // iou: descending merge
                bool ta = a0[pa] >= b0[pb];
                r0[o] = ta ? a0[pa++] : b0[pb++];
            }
            pa = 0; pb = 0;
            for (int o = 0; o < 4; o++) {            // cost: ascending, idx tie-break
                bool ta = (a1[pa] < b1[pb]) || (a1[pa] == b1[pb] && ai[pa] <= bi[pb]);
                if (ta) { r1[o] = a1[pa]; ri[o] = ai[pa]; pa++; }
                else    { r1[o] = b1[pb]; ri[o] = bi[pb]; pb++; }
            }
            for (int j = 0; j < 4; j++) { sli[tid][j] = r0[j]; slc[tid][j] = r1[j]; sci[tid][j] = ri[j]; }
        }
        __syncthreads();
    }
    if (tid == 0) {
        float ssum = 0.0f;
        for (int j = 0; j < 4; j++) ssum += fmaxf(sli[0][j], 0.0f);
        int k = (int)ssum;                // trunc toward zero (sum >= 0)
        if (k < 1) k = 1;
        if (k > 4) k = 4;
        for (int j = 0; j < k; j++) {
            int n = sci[0][j];
            if (n >= 0 && n < NN) atomicOr(&MATCH[b * NN + n], 1 << m);
        }
    }
}

// ---------------- pass D: conflict resolution + outputs
__global__ void __launch_bounds__(256) final_kernel(
    const int* __restrict__ MATCH, const float* __restrict__ COST,
    float* __restrict__ out) {
    int b = blockIdx.y;
    int n = blockIdx.x * 256 + threadIdx.x;
    int w = MATCH[b * NN + n];
    int cnt = __popc(w);
    float asn = 0.0f, mt = -1.0f;
    if (cnt == 1) {
        asn = 1.0f;
        mt = (float)(__ffs(w) - 1);
    } else if (cnt > 1) {
        asn = 1.0f;
        float best = 3.4e38f; int bmi = -1;
        for (int m = 0; m < MM; m++) {
            if ((w >> m) & 1) {
                float c = COST[(size_t)(b * MM + m) * NN + n];
                if (c < best) { best = c; bmi = m; }   // strict '<' keeps first min
            }
        }
        mt = (float)bmi;
    }
    out[b * NN + n] = asn;
    out[BB * NN + b * NN + n] = mt;
}

extern "C" void kernel_launch(void* const* d_in, const int* in_sizes, int n_in,
                              void* d_out, int out_size, void* d_ws, size_t ws_size,
                              hipStream_t stream) {
    const float* preds = (const float*)d_in[0];
    const float* tgt   = (const float*)d_in[1];
    const int*   masks = (const int*)d_in[2];
    const int*   imgw  = (const int*)d_in[3];
    // d_in[4] = img_h (unused)

    size_t BMN = (size_t)BB * MM * NN;
    float* S     = (float*)d_ws;
    float* XYSQ  = S + BMN;
    float* COST  = XYSQ + BMN;
    float* IOUW  = COST + BMN;
    int*   LEN   = (int*)(IOUW + BMN);
    int*   MAXD  = LEN + BB * MM;
    int*   MAXXY = MAXD + BB;
    int*   MAXTH = MAXXY + BB;
    int*   MATCH = MAXTH + BB;
    float* out   = (float*)d_out;

    init_kernel<<<(BB * NN + 255) / 256, 256, 0, stream>>>(tgt, imgw, LEN, MAXD, MAXXY, MAXTH, MATCH);
    passA_kernel<<<dim3(NN / 64, BB), 64, 0, stream>>>(preds, tgt, LEN, imgw, S, XYSQ, MAXD, MAXXY, MAXTH);
    passB_kernel<<<dim3(NN / 256, BB), 256, 0, stream>>>(preds, tgt, masks, LEN, MAXD, MAXXY, MAXTH,
                                                         S, XYSQ, COST, IOUW);
    topk_kernel<<<BB * MM, 128, 0, stream>>>(masks, COST, IOUW, MATCH);
    final_kernel<<<dim3(NN / 256, BB), 256, 0, stream>>>(MATCH, COST, out);
}